// KF_flow_12610023981406
// MI455X (gfx1250) — compile-verified
//
#include <hip/hip_runtime.h>
#include <cstdint>
#include <cstddef>

#ifndef __has_builtin
#define __has_builtin(x) 0
#endif

// ---------------------------------------------------------------------------
// Problem constants (match the reference: B=4096, T=2048, 3 inputs / 4 outputs
// per (b,t)).
// ---------------------------------------------------------------------------
constexpr int kB = 4096;
constexpr int kT = 2048;
constexpr int kTT = 32;                    // timesteps per staged chunk
constexpr int kChunks = kT / kTT;          // 64
constexpr int kRows = 32;                  // batches per block == wave32
constexpr int kInPitch4 = 25;              // float4 per input LDS row (24 data + 1 pad)
constexpr int kOutPitch4 = 33;             // float4 per output LDS row (32 data + 1 pad)
constexpr int kLoadsPerChunk = (kRows * kTT * 3) / 4 / 32;   // 24 b128 per chunk

static_assert(kLoadsPerChunk == 24, "chunk sizing");
static_assert(2 * kLoadsPerChunk <= 63, "ASYNCcnt is 6-bit; keep <=63 in flight");

// ---------------------------------------------------------------------------
// CDNA5 async global->LDS copy (ASYNCcnt-tracked). The clang builtin takes
// typed v4i pointers: (v4i AS1* gaddr, v4i AS3* lds, imm offset, imm cpol),
// per the round-1 diagnostic. Inline-asm fallback kept for other toolchains.
// ---------------------------------------------------------------------------
typedef int v4i __attribute__((vector_size(16)));
typedef __attribute__((address_space(1))) v4i gv4i;
typedef __attribute__((address_space(3))) v4i lv4i;

__device__ __forceinline__ void async_copy_b128(const float4* gp, float4* lp) {
#if __has_builtin(__builtin_amdgcn_global_load_async_to_lds_b128)
  __builtin_amdgcn_global_load_async_to_lds_b128((gv4i*)gp, (lv4i*)lp, 0, 0);
#else
  unsigned lofs = (unsigned)(unsigned long long)
      (__attribute__((address_space(3))) void*)lp;           // LDS byte offset
  unsigned long long ga = (unsigned long long)gp;            // global VA
  asm volatile("global_load_async_to_lds_b128 %0, %1, off"
               :: "v"(lofs), "v"(ga) : "memory");
#endif
}

template <int N>
__device__ __forceinline__ void wait_async() {
#if __has_builtin(__builtin_amdgcn_s_wait_asynccnt)
  __builtin_amdgcn_s_wait_asynccnt(N);
#else
  asm volatile("s_wait_asynccnt %0" :: "i"(N) : "memory");
#endif
}

// ---------------------------------------------------------------------------
// The full pipeline, templated on whether F has the canonical
// [[1, eps], [0, 1]] structure (saves ~11 VALU ops/step on the serial chain).
// ---------------------------------------------------------------------------
template <bool CANON>
__device__ __forceinline__ void kf_pipeline(
    const float4* __restrict__ y4, float4* __restrict__ out4,
    float4* s_in, float4* s_out,
    float f00, float f01, float f10, float f11,
    float q2, float r2,
    float x0, float x1, float p00, float p01, float p11) {
  const int lane = threadIdx.x;
  const int b0 = blockIdx.x * kRows;

  // Stage chunk k of y for all 32 batch rows: each row is 96 contiguous dwords
  // (24 float4) in global, rows strided by T*3 floats. 24 async b128 ops.
  auto issue = [&](int k, int buf) {
    float4* lbase = s_in + (size_t)buf * (kRows * kInPitch4);
#pragma unroll
    for (int it = 0; it < kLoadsPerChunk; ++it) {
      int u = it * 32 + lane;      // 0..767
      int r = u / 24;              // batch row 0..31
      int c = u - r * 24;          // float4 within row 0..23
      const float4* gp = y4 + (size_t)(b0 + r) * (kT * 3 / 4)
                            + (size_t)k * (kTT * 3 / 4) + c;
      async_copy_b128(gp, lbase + r * kInPitch4 + c);
    }
  };

  issue(0, 0);
  issue(1, 1);

  for (int k = 0; k < kChunks; ++k) {
    const int buf = k % 3;
    if (k + 2 < kChunks) {
      issue(k + 2, (k + 2) % 3);
      wait_async<2 * kLoadsPerChunk>();   // loads retire in order: chunk k done
    } else if (k + 1 < kChunks) {
      wait_async<kLoadsPerChunk>();
    } else {
      wait_async<0>();
    }
    __syncthreads();   // single-wave WG: S_NOP + compiler memory fence

    const float* row = (const float*)s_in
                     + (size_t)buf * (kRows * kInPitch4 * 4)
                     + (size_t)lane * (kInPitch4 * 4);
#pragma unroll
    for (int tt = 0; tt < kTT; ++tt) {
      float ym = row[tt * 3 + 0];
      float h0 = row[tt * 3 + 1];
      float h1 = row[tt * 3 + 2];
      float a0, a1, q00, q01, q11;
      if (CANON) {                       // F = [[1, f01], [0, 1]]
        a0 = fmaf(f01, x1, x0);
        a1 = x1;
        float t00 = fmaf(f01, p01, p00);
        float t01 = fmaf(f01, p11, p01);
        q00 = fmaf(f01, t01, t00) + q2;
        q01 = t01;
        q11 = p11 + q2;
      } else {                           // general F: P' = F P F^T + q^2 I
        a0 = fmaf(f00, x0, f01 * x1);
        a1 = fmaf(f10, x0, f11 * x1);
        float t00 = fmaf(f00, p00, f01 * p01);
        float t01 = fmaf(f00, p01, f01 * p11);
        float t10 = fmaf(f10, p00, f11 * p01);
        float t11 = fmaf(f10, p01, f11 * p11);
        q00 = fmaf(f00, t00, f01 * t01) + q2;
        q01 = fmaf(f10, t00, f11 * t01);
        q11 = fmaf(f10, t10, f11 * t11) + q2;
      }
      float u0 = fmaf(q00, h0, q01 * h1);          // P' H^T
      float u1 = fmaf(q01, h0, q11 * h1);
      float s  = fmaf(h0, u0, fmaf(h1, u1, r2));   // innovation variance
      float inv = __builtin_amdgcn_rcpf(s);
      inv = fmaf(fmaf(-s, inv, 1.0f), inv, inv);   // one Newton step
      float dy = ym - fmaf(h0, a0, h1 * a1);
      float kg0 = u0 * inv, kg1 = u1 * inv;        // Kalman gain
      x0 = fmaf(kg0, dy, a0);
      x1 = fmaf(kg1, dy, a1);
      p00 = fmaf(-kg0, u0, q00);                   // P' - K S K^T = P' - u u^T/S
      p01 = fmaf(-kg0, u1, q01);
      p11 = fmaf(-kg1, u1, q11);
      s_out[lane * kOutPitch4 + tt] = make_float4(dy, x0, x1, s);
    }
    __syncthreads();

    // Coalesced write-out: per batch row, 32 lanes store 512 contiguous bytes.
#pragma unroll
    for (int bl = 0; bl < kRows; ++bl) {
      float4 v = s_out[bl * kOutPitch4 + lane];
      out4[(size_t)(b0 + bl) * kT + (size_t)k * kTT + lane] = v;
    }
    __syncthreads();
  }
}

__global__ __launch_bounds__(32)
void kf_scan_kernel(const float* __restrict__ y, const float* __restrict__ F,
                    const float* __restrict__ m1x0, const float* __restrict__ m2x0,
                    const float* __restrict__ q, const float* __restrict__ r,
                    float* __restrict__ out) {
  __shared__ float4 s_in[3 * kRows * kInPitch4];   // 37.5 KB, triple-buffered
  __shared__ float4 s_out[kRows * kOutPitch4];     // 16.5 KB

  const int b = blockIdx.x * kRows + threadIdx.x;

  float f00 = F[0], f01 = F[1], f10 = F[2], f11 = F[3];
  float qv = q[0], rv = r[0];
  float q2 = qv * qv, r2 = rv * rv;
  float p00 = m2x0[0];
  float p01 = 0.5f * (m2x0[1] + m2x0[2]);          // symmetric by construction
  float p11 = m2x0[3];
  float x0 = m1x0[b * 2 + 0];
  float x1 = m1x0[b * 2 + 1];

  if (f10 == 0.0f && f00 == 1.0f && f11 == 1.0f) { // uniform across block
    kf_pipeline<true>((const float4*)y, (float4*)out, s_in, s_out,
                      f00, f01, f10, f11, q2, r2, x0, x1, p00, p01, p11);
  } else {
    kf_pipeline<false>((const float4*)y, (float4*)out, s_in, s_out,
                       f00, f01, f10, f11, q2, r2, x0, x1, p00, p01, p11);
  }
}

extern "C" void kernel_launch(void* const* d_in, const int* in_sizes, int n_in,
                              void* d_out, int out_size, void* d_ws, size_t ws_size,
                              hipStream_t stream) {
  (void)in_sizes; (void)n_in; (void)out_size; (void)d_ws; (void)ws_size;
  const float* y    = (const float*)d_in[0];
  const float* F    = (const float*)d_in[1];
  const float* m1x0 = (const float*)d_in[2];
  const float* m2x0 = (const float*)d_in[3];
  const float* q    = (const float*)d_in[4];
  const float* r    = (const float*)d_in[5];
  float* out = (float*)d_out;

  dim3 grid(kB / kRows);   // 128 blocks, one wave32 each
  dim3 block(32);
  hipLaunchKernelGGL(kf_scan_kernel, grid, block, 0, stream,
                     y, F, m1x0, m2x0, q, r, out);
}